// HierarchicalCentroidRegularizer_73005854097881
// MI455X (gfx1250) — compile-verified
//
#include <hip/hip_runtime.h>
#include <hip/hip_bf16.h>

#define D 512
#define NFINE 100
#define NSUPER 20
#define DSPLIT 4
#define DCHUNK (D / DSPLIT)   // 128 columns per block
#define K1_THREADS 64

typedef __attribute__((ext_vector_type(2))) float v2f;
typedef __attribute__((ext_vector_type(8))) float v8f;

// ---------------------------------------------------------------------------
// Kernel 1: streaming segment-sum. Block (b,h) handles row range b and column
// chunk h (128 of 512 dims). LDS accumulator [100][128] = 50 KB. Row index is
// block-uniform -> label is a scalar load. Each thread always owns the same
// 2-float column slice -> LDS adds are race-free and deterministic.
// Counts are accumulated in a separate thread-parallel pass (adds of 1.0f are
// exact integers in f32 -> order-independent, still deterministic).
// ---------------------------------------------------------------------------
__global__ __launch_bounds__(K1_THREADS)
void k1_segsum(const float* __restrict__ emb, const int* __restrict__ labels,
               float* __restrict__ pSum, float* __restrict__ pCnt,
               int N, int rowsPerBlock) {
  __shared__ __align__(16) float acc[NFINE * DCHUNK];
  __shared__ float cnt[NFINE];
  const int tid = threadIdx.x;
  const int b = blockIdx.x >> 2;
  const int h = blockIdx.x & 3;

  for (int i = tid; i < NFINE * DCHUNK; i += K1_THREADS) acc[i] = 0.f;
  for (int i = tid; i < NFINE; i += K1_THREADS) cnt[i] = 0.f;
  __syncthreads();

  int r0 = b * rowsPerBlock;
  int r1 = r0 + rowsPerBlock;
  if (r1 > N) r1 = N;
  const size_t colBase = (size_t)h * DCHUNK + 2 * tid;

  for (int row = r0; row < r1; ++row) {
    const int lab = labels[row];                       // block-uniform -> scalar load
    const float2 v = *(const float2*)(emb + (size_t)row * D + colBase);
    __builtin_prefetch(emb + (size_t)(row + 8) * D + colBase, 0, 1); // global_prefetch_b8
    atomicAdd(&acc[lab * DCHUNK + 2 * tid],     v.x);  // ds_add_f32 (no contention)
    atomicAdd(&acc[lab * DCHUNK + 2 * tid + 1], v.y);
  }
  if (h == 0) {
    // thread-parallel count pass (exact integer f32 adds -> deterministic)
    for (int row = r0 + tid; row < r1; row += K1_THREADS)
      atomicAdd(&cnt[labels[row]], 1.f);
  }
  __syncthreads();

  for (int i = tid; i < NFINE * DCHUNK; i += K1_THREADS) {
    const int f = i / DCHUNK, dl = i - f * DCHUNK;
    pSum[((size_t)b * NFINE + f) * D + (size_t)h * DCHUNK + dl] = acc[i];
  }
  if (h == 0)
    for (int f = tid; f < NFINE; f += K1_THREADS) pCnt[b * NFINE + f] = cnt[f];
}

// ---------------------------------------------------------------------------
// Kernel 2: one block per fine class. Fixed-order reduction over B partials,
// writes fine_sum / fine_cnt and the per-class fine loss term.
// ---------------------------------------------------------------------------
__global__ __launch_bounds__(256)
void k2_reduce(const float* __restrict__ pSum, const float* __restrict__ pCnt,
               const float* __restrict__ refFine,
               float* __restrict__ fineSum, float* __restrict__ fineCnt,
               float* __restrict__ fineLossPer, int B) {
  __shared__ float red[256];
  __shared__ float cntSh;
  const int f = blockIdx.x, tid = threadIdx.x;

  float s0 = 0.f, s1 = 0.f;
  for (int b = 0; b < B; ++b) {
    const float2 v = *(const float2*)(pSum + ((size_t)b * NFINE + f) * D + 2 * tid);
    s0 += v.x; s1 += v.y;
  }
  *(float2*)(fineSum + (size_t)f * D + 2 * tid) = make_float2(s0, s1);

  if (tid == 0) {
    float c = 0.f;
    for (int b = 0; b < B; ++b) c += pCnt[b * NFINE + f];
    fineCnt[f] = c; cntSh = c;
  }
  __syncthreads();
  const float c = cntSh;
  const float inv = 1.f / fmaxf(c, 1.f);
  const float d0 = s0 * inv - refFine[(size_t)f * D + 2 * tid];
  const float d1 = s1 * inv - refFine[(size_t)f * D + 2 * tid + 1];
  red[tid] = d0 * d0 + d1 * d1;
  __syncthreads();
  for (int s = 128; s > 0; s >>= 1) { if (tid < s) red[tid] += red[tid + s]; __syncthreads(); }
  if (tid == 0) fineLossPer[f] = (c > 0.f) ? red[0] * (1.f / D) : 0.f;
}

// ---------------------------------------------------------------------------
// Kernel 3: single block. Super centroids, super loss, WMMA Gram (20x20 padded
// to 2x2 tiles of 16x16 via v_wmma_f32_16x16x4_f32), inter loss, final scalar.
// scent carries one extra zero row (row 20) so out-of-range M/N lanes read
// zeros with unconditional ds_load_b64 (no EXEC juggling in the k-loop).
// ---------------------------------------------------------------------------
__global__ __launch_bounds__(256)
void k3_final(const float* __restrict__ fineSum, const float* __restrict__ fineCnt,
              const float* __restrict__ fineLossPer,
              const float* __restrict__ refSuper, const float* __restrict__ refInter,
              float* __restrict__ out) {
  __shared__ __align__(16) float scent[(NSUPER + 1) * D];  // +1 zero pad row, ~42 KB
  __shared__ float gram[32 * 32];
  __shared__ float scnt[NSUPER];
  __shared__ float red[256];
  __shared__ float parts[2];
  const int tid = threadIdx.x;

  if (tid < NSUPER) {
    float c = 0.f;
    for (int ff = 0; ff < 5; ++ff) c += fineCnt[tid * 5 + ff];
    scnt[tid] = c;
  }
  for (int i = tid; i < NSUPER * D; i += 256) {
    const int s = i >> 9, d = i & (D - 1);
    float v = 0.f;
    for (int ff = 0; ff < 5; ++ff) v += fineSum[(size_t)(s * 5 + ff) * D + d];
    scent[i] = v;
  }
  for (int i = tid; i < D; i += 256) scent[NSUPER * D + i] = 0.f;  // pad row
  __syncthreads();
  for (int i = tid; i < NSUPER * D; i += 256) {
    const int s = i >> 9;
    scent[i] = scent[i] / fmaxf(scnt[s], 1.f);
  }
  __syncthreads();

  // super loss
  float sl = 0.f;
  for (int i = tid; i < NSUPER * D; i += 256) {
    const int s = i >> 9;
    if (scnt[s] > 0.f) { const float dd = scent[i] - refSuper[i]; sl += dd * dd; }
  }
  red[tid] = sl * (1.f / D);
  __syncthreads();
  for (int s = 128; s > 0; s >>= 1) { if (tid < s) red[tid] += red[tid + s]; __syncthreads(); }
  if (tid == 0) parts[0] = red[0];
  __syncthreads();

  // Gram = S * S^T via WMMA f32 16x16x4 (K-loop over 512). Waves 0..3 own the
  // four 16x16 tiles of the zero-padded 32x32 result.
  const int wave = tid >> 5;
  const int lane = tid & 31;
  if (wave < 4) {
    const int mT = wave >> 1, nT = wave & 1;
    const int li = lane & 15;
    const int hi = (lane >> 4) & 1;            // lanes 16..31 hold K+2,K+3
    const int m = mT * 16 + li;
    const int n = nT * 16 + li;
    const int mc = (m < NSUPER) ? m : NSUPER;  // row 20 = zero pad
    const int nc = (n < NSUPER) ? n : NSUPER;
    const float* arow = scent + mc * D + 2 * hi;
    const float* brow = scent + nc * D + 2 * hi;
    v8f c = {};
#pragma unroll 4
    for (int k = 0; k < D; k += 4) {
      const v2f a  = *(const v2f*)(arow + k);  // ds_load_b64
      const v2f bb = *(const v2f*)(brow + k);  // ds_load_b64
      c = __builtin_amdgcn_wmma_f32_16x16x4_f32(false, a, false, bb,
                                                (short)0, c, false, false);
    }
    // C layout: VGPR r -> M = r (lanes 0-15) / M = 8+r (lanes 16-31), N = lane&15
    for (int r = 0; r < 8; ++r) {
      const int row = mT * 16 + r + 8 * hi;
      const int col = nT * 16 + li;
      gram[row * 32 + col] = c[r];
    }
  }
  __syncthreads();

  // inter loss over upper triangle, dist^2 = G[ii]+G[jj]-2G[ij]
  float il = 0.f;
  for (int p = tid; p < NSUPER * NSUPER; p += 256) {
    const int i = p / NSUPER, j = p % NSUPER;
    if (i < j && scnt[i] > 0.f && scnt[j] > 0.f) {
      const float sq = gram[i * 32 + i] + gram[j * 32 + j] - 2.f * gram[i * 32 + j];
      const float dist = sqrtf(fmaxf(sq, 0.f));
      const float dd = dist - refInter[i * NSUPER + j];
      il += dd * dd;
    }
  }
  red[tid] = il;
  __syncthreads();
  for (int s = 128; s > 0; s >>= 1) { if (tid < s) red[tid] += red[tid + s]; __syncthreads(); }
  if (tid == 0) parts[1] = red[0];
  __syncthreads();

  float fl = 0.f;
  for (int f = tid; f < NFINE; f += 256) fl += fineLossPer[f];
  red[tid] = fl;
  __syncthreads();
  for (int s = 128; s > 0; s >>= 1) { if (tid < s) red[tid] += red[tid + s]; __syncthreads(); }
  if (tid == 0) out[0] = red[0] + parts[0] + parts[1];
}

// ---------------------------------------------------------------------------
extern "C" void kernel_launch(void* const* d_in, const int* in_sizes, int n_in,
                              void* d_out, int out_size, void* d_ws, size_t ws_size,
                              hipStream_t stream) {
  const float* emb     = (const float*)d_in[0];
  const int*   labels  = (const int*)d_in[1];
  const float* refFine = (const float*)d_in[2];
  const float* refSup  = (const float*)d_in[3];
  const float* refInt  = (const float*)d_in[4];
  const int N = in_sizes[1];

  // Workspace layout: [B][100][512] partial sums, [B][100] partial counts,
  // then fine_sum[100][512], fine_cnt[100], fine_loss_per[100].
  const size_t perB  = (size_t)NFINE * D * sizeof(float) + (size_t)NFINE * sizeof(float);
  const size_t fixed = (size_t)NFINE * D * sizeof(float) + 2 * (size_t)NFINE * sizeof(float) + 256;
  int B = 256;
  while (B > 8 && fixed + (size_t)B * perB > ws_size) B >>= 1;

  char* w = (char*)d_ws;
  float* pSum = (float*)w;        w += (size_t)B * NFINE * D * sizeof(float);
  float* pCnt = (float*)w;        w += (size_t)B * NFINE * sizeof(float);
  float* fineSum = (float*)w;     w += (size_t)NFINE * D * sizeof(float);
  float* fineCnt = (float*)w;     w += NFINE * sizeof(float);
  float* fineLossPer = (float*)w; w += NFINE * sizeof(float);

  const int rowsPerBlock = (N + B - 1) / B;
  k1_segsum<<<B * DSPLIT, K1_THREADS, 0, stream>>>(emb, labels, pSum, pCnt, N, rowsPerBlock);
  k2_reduce<<<NFINE, 256, 0, stream>>>(pSum, pCnt, refFine, fineSum, fineCnt, fineLossPer, B);
  k3_final<<<1, 256, 0, stream>>>(fineSum, fineCnt, fineLossPer, refSup, refInt, (float*)d_out);
}